// KANSplineLayer_75960791597722
// MI455X (gfx1250) — compile-verified
//
#include <hip/hip_runtime.h>

typedef float v2f __attribute__((ext_vector_type(2)));
typedef float v8f __attribute__((ext_vector_type(8)));

#define B_    16
#define IN_   1024
#define OUT_  1024
#define G_    8
#define KCH   8              // K / i chunks for occupancy
#define KLEN  (IN_ / KCH)    // 128

// ---- hardware tanh (1 TRANS op; sigmoid(z) = 0.5 + 0.5*tanh(z/2)) ----
#if defined(__AMDGCN__)
#if __has_builtin(__builtin_amdgcn_tanhf)
__device__ __forceinline__ float tanh_hw(float v) { return __builtin_amdgcn_tanhf(v); }
#else
__device__ __forceinline__ float tanh_hw(float v) {
  float r;
  asm("v_tanh_f32 %0, %1" : "=v"(r) : "v"(v));
  return r;
}
#endif
#else
__device__ __forceinline__ float tanh_hw(float v) { return v; } // host-pass stub, never executed
#endif

__global__ void kan_zero_out(float* __restrict__ out) {
  int t = blockIdx.x * blockDim.x + threadIdx.x;
  if (t < B_ * OUT_) out[t] = 0.0f;
}

// One block = full 16(b) x 16(o) output tile, over one K-chunk of i.
// Wave 0 additionally computes the fp32 WMMA base-GEMM partial for the tile.
__global__ __launch_bounds__(256) void kan_spline_wmma_kernel(
    const float* __restrict__ x,      // [16, 1024]
    const float* __restrict__ W,      // [1024, 1024] (OUT, IN)
    const float* __restrict__ grid,   // [1024, 1024, 8] (IN, OUT, G)
    float* __restrict__ out)          // [16, 1024]
{
  __shared__ float tile[16][17];      // base-GEMM partial, padded vs bank conflicts

  const int ot      = blockIdx.x;          // 0..63  : 16-wide OUT tile
  const int kc      = blockIdx.y;          // 0..7   : K / i chunk
  const int tid     = threadIdx.x;
  const int o_local = tid & 15;
  const int b       = tid >> 4;
  const int o       = ot * 16 + o_local;
  const int k0      = kc * KLEN;

  // ---------------- wave 0: V_WMMA_F32_16X16X4_F32 base GEMM partial -------
  // A = x (M=16 x K), B[k][n] = W[ot*16+n][k].  Per ISA layout:
  //   lanes 0-15 : M/N = lane, hold K=k, k+1 ; lanes 16-31: M/N = lane-16, K=k+2, k+3
  if (tid < 32) {
    const int lane = tid;
    const int hi   = lane >> 4;            // 0 or 1 -> K offset 0 or 2
    const int mn   = lane & 15;            // A row (M) / B col (N)
    v8f acc = {};
    const float* aptr = x + mn * IN_           + k0 + hi * 2;
    const float* bptr = W + (ot * 16 + mn) * IN_ + k0 + hi * 2;
    #pragma unroll 4
    for (int k = 0; k < KLEN; k += 4) {
      v2f a  = *(const v2f*)(aptr + k);
      v2f bb = *(const v2f*)(bptr + k);
      // 8 args: (neg_a, A, neg_b, B, c_mod, C, reuse_a, reuse_b)
      acc = __builtin_amdgcn_wmma_f32_16x16x4_f32(
          false, a, false, bb, (short)0, acc, false, false);
    }
    // C/D layout: VGPR r, lanes 0-15 -> M=r, N=lane ; lanes 16-31 -> M=r+8
    #pragma unroll
    for (int r = 0; r < 8; ++r)
      tile[r + hi * 8][mn] = acc[r];
  }
  __syncthreads();

  // ---------------- all 256 threads: spline partial over i in [k0, k0+KLEN)
  // sigmoid(x+g) = 0.5 + 0.5*tanh(0.5*(x+g)); constant folded analytically.
  const float* gp = grid + ((size_t)k0 * OUT_ + o) * G_;   // 32B aligned
  const size_t gstep = (size_t)OUT_ * G_;
  float s = 0.0f;
  for (int i = 0; i < KLEN; ++i) {
    const float hx = 0.5f * x[b * IN_ + k0 + i];
    const float4 g0 = *(const float4*)(gp);
    const float4 g1 = *(const float4*)(gp + 4);
#if defined(__AMDGCN__)
    __builtin_prefetch(gp + 8 * gstep, 0, 1);   // global_prefetch_b8, 8 rows ahead
#endif
    gp += gstep;
    s += tanh_hw(__builtin_fmaf(0.5f, g0.x, hx));
    s += tanh_hw(__builtin_fmaf(0.5f, g0.y, hx));
    s += tanh_hw(__builtin_fmaf(0.5f, g0.z, hx));
    s += tanh_hw(__builtin_fmaf(0.5f, g0.w, hx));
    s += tanh_hw(__builtin_fmaf(0.5f, g1.x, hx));
    s += tanh_hw(__builtin_fmaf(0.5f, g1.y, hx));
    s += tanh_hw(__builtin_fmaf(0.5f, g1.z, hx));
    s += tanh_hw(__builtin_fmaf(0.5f, g1.w, hx));
  }

  // per-chunk share of the analytic constant: (IN_*G_*0.5)/KCH = 512
  const float val = tile[b][o_local] + 0.5f * s + (float)(IN_ * G_) * 0.5f / KCH;
  atomicAdd(&out[b * OUT_ + o], val);
}

extern "C" void kernel_launch(void* const* d_in, const int* in_sizes, int n_in,
                              void* d_out, int out_size, void* d_ws, size_t ws_size,
                              hipStream_t stream) {
  const float* x    = (const float*)d_in[0];   // [16,1024]
  const float* W    = (const float*)d_in[1];   // [1024,1024]
  const float* grid = (const float*)d_in[2];   // [1024,1024,8]
  float*       out  = (float*)d_out;           // [16,1024]

  kan_zero_out<<<(B_ * OUT_ + 255) / 256, 256, 0, stream>>>(out);
  kan_spline_wmma_kernel<<<dim3(OUT_ / 16, KCH), 256, 0, stream>>>(x, W, grid, out);
}